// Linear_90701119357381
// MI455X (gfx1250) — compile-verified
//
#include <hip/hip_runtime.h>

// ---------------------------------------------------------------------------
// Problem constants (match reference)
// ---------------------------------------------------------------------------
#define N_NODES 10000
#define N_EDGES 320000
#define N_GSO   2
#define BATCH   4
#define F0      64
#define F1      128
#define F2      128
#define M_ROWS  (N_NODES * BATCH)   // 40000 rows for the dense GEMMs
#define FOUT    128                 // both layers have Fout = 128
// LDS W layout: k-pair interleaved float2 cells, row stride S2 cells.
// 2*S2 mod 64 == 32 -> the two half-wave lane groups hit disjoint bank halves.
#define S2      144

typedef float v2f __attribute__((ext_vector_type(2)));
typedef float v8f __attribute__((ext_vector_type(8)));

// ---------------------------------------------------------------------------
// Native f32 atomic add (global_atomic_add_f32)
// ---------------------------------------------------------------------------
__device__ __forceinline__ void atomic_add_f32(float* p, float v) {
    unsafeAtomicAdd(p, v);   // lowers to global_atomic_add_f32 on gfx1250
}

// ---------------------------------------------------------------------------
// x [B, N, F0] -> z0 [N, B, F0]   (node-major)
// ---------------------------------------------------------------------------
__global__ void transpose_x_k(const float* __restrict__ x, float* __restrict__ z) {
    int i = blockIdx.x * blockDim.x + threadIdx.x;        // over N*B*F0
    if (i >= N_NODES * BATCH * F0) return;
    int f = i % F0;
    int b = (i / F0) % BATCH;
    int n = i / (F0 * BATCH);
    z[i] = x[((size_t)b * N_NODES + n) * F0 + f];
}

// h[m, f] = bias[f]   (fold bias into the accumulator init)
__global__ void init_bias_k(float* __restrict__ h, const float* __restrict__ bias, int n) {
    int i = blockIdx.x * blockDim.x + threadIdx.x;
    if (i < n) h[i] = bias[i & (FOUT - 1)];
}

__global__ void zero_k(float* __restrict__ p, int n) {
    int i = blockIdx.x * blockDim.x + threadIdx.x;
    if (i < n) p[i] = 0.0f;
}

// out[b*F2+f] = h2[node0, b, f]  (node-major: first BATCH*F2 floats)
__global__ void extract_k(const float* __restrict__ h2, float* __restrict__ out) {
    int i = blockIdx.x * blockDim.x + threadIdx.x;
    if (i < BATCH * F2) out[i] = h2[i];
}

// ---------------------------------------------------------------------------
// SpMM:  zout[rows[e], :] += vals[e] * zin[cols[e], :]   blocks of BF floats
// One wave (32 lanes) per edge; coalesced block gather, f32 atomics scatter.
// ---------------------------------------------------------------------------
template <int BF>
__global__ __launch_bounds__(256) void spmm_k(const int*   __restrict__ rows,
                                              const int*   __restrict__ cols,
                                              const float* __restrict__ vals,
                                              const float* __restrict__ zin,
                                              float*       __restrict__ zout) {
    int gw   = (blockIdx.x * blockDim.x + threadIdx.x) >> 5;  // wave id == edge id
    int lane = threadIdx.x & 31;
    if (gw >= N_EDGES) return;
    int   r = rows[gw];
    int   c = cols[gw];
    float v = vals[gw];
    const float* src = zin  + (size_t)c * BF;
    float*       dst = zout + (size_t)r * BF;
#pragma unroll
    for (int j = lane; j < BF; j += 32) {
        atomic_add_f32(&dst[j], v * src[j]);
    }
}

// ---------------------------------------------------------------------------
// Dense GEMM accumulate:  C[M,128] += A[M,FIN] @ W[FIN,128]
// f32-exact via V_WMMA_F32_16X16X4_F32.
//
// Block = 128 threads = 4 waves; block tile = 64(M) x 128(N).
// W staged once per block into LDS, k-pair interleaved:
//     cell[(k>>1)*S2 + col] = { W[k][col], W[k+1][col] }      (float2)
// so each B fragment is ONE aligned ds_load_b64 (no register shuffles),
// and the two half-wave lane groups (k-pair kb/2 vs kb/2+1) land on
// disjoint LDS bank halves (2*S2 mod 64 == 32).
//
// Each wave owns a 32(M) x 64(N) tile: 8 accumulators; per k-step:
// 2x global b64 (A) + 4x ds_load_b64 (B) + 8x WMMA.
//
// Fragment layouts (wave32, per ISA 7.12.2):
//   A 16x4 : lane<16 -> row=lane,    {K=kb+0, K=kb+1} in {a.x, a.y}
//            lane>=16 -> row=lane-16, {K=kb+2, K=kb+3}
//   B 4x16 : symmetric (col = lane&15, same K split by lane group)
//   C/D    : vgpr i holds row (i + 8*(lane>=16)), col = lane&15
// ---------------------------------------------------------------------------
template <int FIN>
__global__ __launch_bounds__(128) void gemm_acc_k(const float* __restrict__ A,
                                                  const float* __restrict__ W,
                                                  float*       __restrict__ C) {
    __shared__ float Wlds[(FIN / 2) * S2 * 2];            // float2 cells, flat as float

    const int tid  = threadIdx.x;
    const int lane = tid & 31;
    const int wid  = tid >> 5;                            // 0..3 waves per block

    // ---- cooperative stage of W[FIN,128] into k-pair-interleaved LDS ----
#pragma unroll
    for (int idx = tid; idx < FIN * FOUT; idx += 128) {
        int k   = idx >> 7;          // source row
        int col = idx & 127;         // source column
        Wlds[((k >> 1) * S2 + col) * 2 + (k & 1)] = W[idx];
    }
    __syncthreads();

    // ---- wave tile mapping: 2 m-pairs x 2 n-quads per block ----
    const int m_pair = blockIdx.x * 2 + (wid >> 1);       // 0..1249 (32-row tiles)
    const int n_quad = wid & 1;                           // 0..1    (64-col quads)

    const int r  = lane & 15;                             // row/col within tile
    const int kh = lane >> 4;                             // K half-select

    const float* Arow0 = A + (size_t)(m_pair * 32 + r)      * FIN + 2 * kh;
    const float* Arow1 = A + (size_t)(m_pair * 32 + 16 + r) * FIN + 2 * kh;
    const int    colbase = n_quad * 64 + r;

    v8f acc[2][4] = {};                                   // [m-sub][n-tile]

#pragma unroll 4
    for (int kb = 0; kb < FIN; kb += 4) {
        v2f a0 = *(const v2f*)(Arow0 + kb);
        v2f a1 = *(const v2f*)(Arow1 + kb);
        // B fragment cells: k-pair row (kb/2 + kh), cols colbase + t*16
        const float* Lb = &Wlds[(((kb >> 1) + kh) * S2 + colbase) * 2];
#pragma unroll
        for (int t = 0; t < 4; ++t) {
            v2f b = *(const v2f*)(Lb + t * 32);           // one ds_load_b64
            acc[0][t] = __builtin_amdgcn_wmma_f32_16x16x4_f32(
                false, a0, false, b, (short)0, acc[0][t], false, false);
            acc[1][t] = __builtin_amdgcn_wmma_f32_16x16x4_f32(
                false, a1, false, b, (short)0, acc[1][t], false, false);
        }
    }

    // ---- accumulate into C (tiles exclusively owned by this wave) ----
#pragma unroll
    for (int ms = 0; ms < 2; ++ms) {
#pragma unroll
        for (int t = 0; t < 4; ++t) {
            const int col = n_quad * 64 + t * 16 + r;
#pragma unroll
            for (int i = 0; i < 8; ++i) {
                const int row = m_pair * 32 + ms * 16 + kh * 8 + i;
                C[(size_t)row * FOUT + col] += acc[ms][t][i];
            }
        }
    }
}

// ---------------------------------------------------------------------------
// Orchestration
// ---------------------------------------------------------------------------
extern "C" void kernel_launch(void* const* d_in, const int* in_sizes, int n_in,
                              void* d_out, int out_size, void* d_ws, size_t ws_size,
                              hipStream_t stream) {
    const float* x   = (const float*)d_in[0];
    const int*   ei  = (const int*)  d_in[1];   // [N_GSO, 2, E]
    const float* ev  = (const float*)d_in[2];   // [N_GSO, E]
    const float* W1  = (const float*)d_in[3];   // [2,3,64,128]
    const float* b1  = (const float*)d_in[4];   // [128]
    const float* W2  = (const float*)d_in[5];   // [2,3,128,128]
    const float* b2  = (const float*)d_in[6];   // [128]
    float*       out = (float*)d_out;           // [4,128]

    const size_t SLOT = (size_t)M_ROWS * 128;   // floats per slot (20.48 MB)
    float* s0 = (float*)d_ws;                   // z0 (layer1 input) -> later h2
    float* s1 = s0 + SLOT;                      // zA (diffusion ping)
    float* s2 = s1 + SLOT;                      // zB (diffusion pong)
    float* s3 = s2 + SLOT;                      // h1 (layer1 output / layer2 input)

    const dim3 B256(256);
    const dim3 B128(128);
    const dim3 G_trans((N_NODES * BATCH * F0 + 255) / 256);      // 10000
    const dim3 G_full((M_ROWS * FOUT + 255) / 256);              // 20000
    const dim3 G_half((M_ROWS * F0 + 255) / 256);                // 10000
    const dim3 G_spmm((N_EDGES * 32 + 255) / 256);               // 40000
    const dim3 G_gemm(M_ROWS / 64);                              // 625 (exact)

    // ---- Layer 1:  h1 = b1 + sum_e sum_k (S_e^k z0) @ W1[e,k],  Fin=64 ----
    transpose_x_k<<<G_trans, B256, 0, stream>>>(x, s0);
    init_bias_k<<<G_full, B256, 0, stream>>>(s3, b1, M_ROWS * FOUT);

    for (int e = 0; e < N_GSO; ++e) {
        const int*   rows = ei + (size_t)e * 2 * N_EDGES;
        const int*   cols = rows + N_EDGES;
        const float* vals = ev + (size_t)e * N_EDGES;
        const float* W1e  = W1 + (size_t)e * 3 * F0 * FOUT;

        gemm_acc_k<F0><<<G_gemm, B128, 0, stream>>>(s0, W1e + 0 * F0 * FOUT, s3);

        zero_k<<<G_half, B256, 0, stream>>>(s1, M_ROWS * F0);
        spmm_k<BATCH * F0><<<G_spmm, B256, 0, stream>>>(rows, cols, vals, s0, s1);
        gemm_acc_k<F0><<<G_gemm, B128, 0, stream>>>(s1, W1e + 1 * F0 * FOUT, s3);

        zero_k<<<G_half, B256, 0, stream>>>(s2, M_ROWS * F0);
        spmm_k<BATCH * F0><<<G_spmm, B256, 0, stream>>>(rows, cols, vals, s1, s2);
        gemm_acc_k<F0><<<G_gemm, B128, 0, stream>>>(s2, W1e + 2 * F0 * FOUT, s3);
    }

    // ---- Layer 2:  h2 = b2 + sum_e sum_k (S_e^k h1) @ W2[e,k],  Fin=128 ----
    init_bias_k<<<G_full, B256, 0, stream>>>(s0, b2, M_ROWS * FOUT);   // s0 now h2

    for (int e = 0; e < N_GSO; ++e) {
        const int*   rows = ei + (size_t)e * 2 * N_EDGES;
        const int*   cols = rows + N_EDGES;
        const float* vals = ev + (size_t)e * N_EDGES;
        const float* W2e  = W2 + (size_t)e * 3 * F1 * FOUT;

        gemm_acc_k<F1><<<G_gemm, B128, 0, stream>>>(s3, W2e + 0 * F1 * FOUT, s0);

        zero_k<<<G_full, B256, 0, stream>>>(s1, M_ROWS * F1);
        spmm_k<BATCH * F1><<<G_spmm, B256, 0, stream>>>(rows, cols, vals, s3, s1);
        gemm_acc_k<F1><<<G_gemm, B128, 0, stream>>>(s1, W2e + 1 * F1 * FOUT, s0);

        zero_k<<<G_full, B256, 0, stream>>>(s2, M_ROWS * F1);
        spmm_k<BATCH * F1><<<G_spmm, B256, 0, stream>>>(rows, cols, vals, s1, s2);
        gemm_acc_k<F1><<<G_gemm, B128, 0, stream>>>(s2, W2e + 2 * F1 * FOUT, s0);
    }

    // ---- Extract node 0: out[b, f] = h2[0, b, f] ----
    extract_k<<<dim3(2), B256, 0, stream>>>(s0, out);
}